// VRNNLoss_46995532153527
// MI455X (gfx1250) — compile-verified
//
#include <hip/hip_runtime.h>

// ---------------------------------------------------------------------------
// VRNN loss -> scalar. Bandwidth-bound streaming reduction (~192 MB read).
// Wave reduction done with V_WMMA_F32_16X16X4_F32 (full-f32 CDNA5 WMMA):
//   A = 16x4 tile = the 64 per-lane partials of one wave32 (2 VGPRs/lane),
//   B = ones     => D[m][n] = row-sum of A. 7 adds + shfl_xor(16) -> wave sum.
// ---------------------------------------------------------------------------

typedef __attribute__((ext_vector_type(2))) float v2f;
typedef __attribute__((ext_vector_type(8))) float v8f;

#define T_DIM 64
#define B_DIM 512
#define Z_DIM 128
#define D_DIM 256
#define LOG_2PI 1.8378770664093453

// ---- wave + block reduction; returns block sum in thread 0 ----------------
__device__ __forceinline__ float block_reduce_wmma(float acc0, float acc1,
                                                   float* warp_sums) {
    v2f a;  a.x = acc0; a.y = acc1;          // this wave's 64 partials = A(16x4)
    v2f bo; bo.x = 1.0f; bo.y = 1.0f;        // B(4x16) = ones
    v8f c = {};
    // (neg_a, A, neg_b, B, c_mod, C, reuse_a, reuse_b)
    v8f d = __builtin_amdgcn_wmma_f32_16x16x4_f32(
        false, a, false, bo, (short)0, c, false, false);
    // lane<16 holds rows 0..7 (one per VGPR), lane>=16 holds rows 8..15
    float s = d[0] + d[1] + d[2] + d[3] + d[4] + d[5] + d[6] + d[7];
    s += __shfl_xor(s, 16, 32);              // lanes 0..31 all get wave total

    const int lane = threadIdx.x & 31;
    const int wave = threadIdx.x >> 5;
    if (lane == 0) warp_sums[wave] = s;
    __syncthreads();

    float bs = 0.0f;
    if (threadIdx.x == 0) {
        const int nw = blockDim.x >> 5;
        for (int i = 0; i < nw; ++i) bs += warp_sums[i];
    }
    return bs;
}

// ---- KLD elementwise: ps - es + exp(es-ps) + (em-pm)^2*exp(-ps) - 1 -------
__device__ __forceinline__ float kld_el(float pm, float ps, float em, float es) {
    float t = em - pm;
    return (ps - es - 1.0f) + expf(es - ps) + t * t * expf(-ps);
}

__global__ __launch_bounds__(256)
void kld_kernel(const float* __restrict__ pm, const float* __restrict__ ps,
                const float* __restrict__ em, const float* __restrict__ es,
                float* __restrict__ partials, int ngroups) {
    __shared__ float warp_sums[8];
    const float4* pm4 = (const float4*)pm;
    const float4* ps4 = (const float4*)ps;
    const float4* em4 = (const float4*)em;
    const float4* es4 = (const float4*)es;

    float acc0 = 0.0f, acc1 = 0.0f;
    const int stride = gridDim.x * blockDim.x;
    for (int g = blockIdx.x * blockDim.x + threadIdx.x; g < ngroups; g += stride) {
        float4 a = pm4[g], b = ps4[g], c = em4[g], d = es4[g];
        acc0 += kld_el(a.x, b.x, c.x, d.x) + kld_el(a.y, b.y, c.y, d.y);
        acc1 += kld_el(a.z, b.z, c.z, d.z) + kld_el(a.w, b.w, c.w, d.w);
    }
    float bs = block_reduce_wmma(acc0, acc1, warp_sums);
    if (threadIdx.x == 0) partials[blockIdx.x] = bs;
}

// ---- NLL elementwise (mask in {0,1}): lv + diff^2 * exp(-lv) --------------
__device__ __forceinline__ float nll_el(float x, float dm, float ds, int mi) {
    float m  = (float)mi;
    float lv = 0.5f * ds * m;
    float df = (x - dm) * m;
    return lv + df * df * expf(-lv);
}

__global__ __launch_bounds__(256)
void nll_kernel(const float* __restrict__ x, const float* __restrict__ dm,
                const float* __restrict__ ds, const int* __restrict__ msk,
                float* __restrict__ partials, int ngroups) {
    __shared__ float warp_sums[8];
    const float4* x4  = (const float4*)x;
    const float4* dm4 = (const float4*)dm;
    const float4* ds4 = (const float4*)ds;
    const int4*   m4  = (const int4*)msk;

    float acc0 = 0.0f, acc1 = 0.0f;
    const int stride = gridDim.x * blockDim.x;
    for (int g = blockIdx.x * blockDim.x + threadIdx.x; g < ngroups; g += stride) {
        // dec tensors are [T,B,D] flat in float4 groups; msk is [B,T,D].
        // B*D/4 = 32768 = 2^15 ; D/4 = 64 = 2^6
        int t   = g >> 15;
        int rem = g & 32767;
        int b   = rem >> 6;
        int dv  = rem & 63;
        int mg  = (b * T_DIM + t) * (D_DIM / 4) + dv;

        float4 vx = x4[g], vdm = dm4[g], vds = ds4[g];
        int4   vm = m4[mg];
        acc0 += nll_el(vx.x, vdm.x, vds.x, vm.x) + nll_el(vx.y, vdm.y, vds.y, vm.y);
        acc1 += nll_el(vx.z, vdm.z, vds.z, vm.z) + nll_el(vx.w, vdm.w, vds.w, vm.w);
    }
    float bs = block_reduce_wmma(acc0, acc1, warp_sums);
    if (threadIdx.x == 0) partials[blockIdx.x] = bs;
}

// ---- deterministic finalize: fixed-order double-precision reduction -------
__global__ __launch_bounds__(256)
void finalize_kernel(const float* __restrict__ partials, int n,
                     float* __restrict__ out) {
    __shared__ double sh[256];
    double s = 0.0;
    for (int i = threadIdx.x; i < n; i += 256) s += (double)partials[i];
    sh[threadIdx.x] = s;
    __syncthreads();
    for (int off = 128; off > 0; off >>= 1) {
        if ((int)threadIdx.x < off) sh[threadIdx.x] += sh[threadIdx.x + off];
        __syncthreads();
    }
    if (threadIdx.x == 0) {
        // loss = (0.5/B)*(S_kld + S_nll) + 0.5*T*D*log(2*pi)
        double loss = (0.5 / (double)B_DIM) * sh[0]
                    + 0.5 * (double)T_DIM * (double)D_DIM * LOG_2PI;
        out[0] = (float)loss;
    }
}

extern "C" void kernel_launch(void* const* d_in, const int* in_sizes, int n_in,
                              void* d_out, int out_size, void* d_ws, size_t ws_size,
                              hipStream_t stream) {
    const float* pm  = (const float*)d_in[0];  // all_prior_mean [T,B,Z]
    const float* ps  = (const float*)d_in[1];  // all_prior_std  [T,B,Z]
    const float* x   = (const float*)d_in[2];  // all_x          [T,B,D]
    const float* em  = (const float*)d_in[3];  // all_enc_mean   [T,B,Z]
    const float* es  = (const float*)d_in[4];  // all_enc_std    [T,B,Z]
    const float* dm  = (const float*)d_in[5];  // all_dec_mean   [T,B,D]
    const float* ds  = (const float*)d_in[6];  // all_dec_std    [T,B,D]
    const int*   msk = (const int*)d_in[7];    // msk            [B,T,D]
    // d_in[8], d_in[9] (eval_x, eval_msk) do not affect the loss: not read.

    float* partials = (float*)d_ws;            // 2048 floats = 8 KB scratch
    const int K1 = 1024, K2 = 1024;
    const int ng_kld = (T_DIM * B_DIM * Z_DIM) / 4;  // 1,048,576 float4 groups
    const int ng_nll = (T_DIM * B_DIM * D_DIM) / 4;  // 2,097,152 float4 groups

    kld_kernel<<<K1, 256, 0, stream>>>(pm, ps, em, es, partials, ng_kld);
    nll_kernel<<<K2, 256, 0, stream>>>(x, dm, ds, msk, partials + K1, ng_nll);
    finalize_kernel<<<1, 256, 0, stream>>>(partials, K1 + K2, (float*)d_out);
}